// SparseMessagePassing_80341658239668
// MI455X (gfx1250) — compile-verified
//
#include <hip/hip_runtime.h>

typedef float v2f __attribute__((ext_vector_type(2)));
typedef float v4f __attribute__((ext_vector_type(4)));
typedef float v8f __attribute__((ext_vector_type(8)));

#define H      128
#define K3     384          // 3*H concatenated input dim
#define TILE_M 64           // edges per block (4 WMMA M sub-tiles)
#define MSUB   (TILE_M / 16)
#define LDA    388          // LDS row stride (floats): 388%64==4 -> conflict-free b64 reads, 16B-aligned rows

// ---------------------------------------------------------------------------
// Kernel 1: edge message GEMM  msg = relu(cat @ W1 + b1), atomically
// scattered into out[receiver] (out pre-seeded with x -> residual included).
// ---------------------------------------------------------------------------
__global__ __launch_bounds__(256)
void edge_msg_kernel(const float* __restrict__ x,
                     const int*   __restrict__ senders,
                     const int*   __restrict__ receivers,
                     const float* __restrict__ ef,
                     const float* __restrict__ W1,   // [384,128] row-major
                     const float* __restrict__ b1,   // [128]
                     float*       __restrict__ out,  // [N,128], pre-seeded with x
                     int nEdges)
{
    __shared__ float As[TILE_M * LDA];   // gathered cat tile: 64 x 384 (padded) ~97 KB
    __shared__ int   sIdx[TILE_M];
    __shared__ int   rIdx[TILE_M];

    const int tid = threadIdx.x;
    const long long e0 = (long long)blockIdx.x * TILE_M;

    if (tid < TILE_M) {
        long long e = e0 + tid;
        int ec = (e < nEdges) ? (int)e : (nEdges - 1);
        sIdx[tid] = senders[ec];
        rIdx[tid] = receivers[ec];
    }
    __syncthreads();

    // Cooperative gather of the A tile: 64 rows x 96 float4 = 6144 loads, 24/thread.
    #pragma unroll
    for (int i = 0; i < (TILE_M * (K3 / 4)) / 256; ++i) {
        int idx = tid + i * 256;
        int row = idx / (K3 / 4);
        int col = (idx % (K3 / 4)) * 4;
        long long e = e0 + row;
        int ec = (e < nEdges) ? (int)e : (nEdges - 1);
        v4f v;
        if (col < H) {
            v = *(const v4f*)(x + (long long)sIdx[row] * H + col);
        } else if (col < 2 * H) {
            v = *(const v4f*)(x + (long long)rIdx[row] * H + (col - H));
        } else {
            v = *(const v4f*)(ef + (long long)ec * H + (col - 2 * H));
        }
        *(v4f*)&As[row * LDA + col] = v;
    }
    __syncthreads();

    // Wave w owns output columns [w*16, w*16+16); four 16-row M sub-tiles.
    const int w    = tid >> 5;          // 0..7
    const int lane = tid & 31;
    const int lrow = lane & 15;         // row/col-in-tile selector
    const int lhi  = lane >> 4;         // K-half selector
    const int col  = (w << 4) + lrow;   // this lane's output column

    v8f acc[MSUB];
    #pragma unroll
    for (int m = 0; m < MSUB; ++m)
        acc[m] = (v8f){0.f, 0.f, 0.f, 0.f, 0.f, 0.f, 0.f, 0.f};

    #pragma unroll 4
    for (int k = 0; k < K3; k += 4) {
        // B fragment: 4x16 slice of W1 (L2-resident), reused by 4 WMMAs.
        // lanes 0-15 hold K=k,k+1; lanes 16-31 hold K=k+2,k+3.
        v2f b;
        b.x = W1[(k + 2 * lhi)     * H + col];
        b.y = W1[(k + 2 * lhi + 1) * H + col];

        #pragma unroll
        for (int m = 0; m < MSUB; ++m) {
            v2f a = *(const v2f*)&As[(m * 16 + lrow) * LDA + k + 2 * lhi];
            acc[m] = __builtin_amdgcn_wmma_f32_16x16x4_f32(false, a, false, b,
                                                           (short)0, acc[m],
                                                           false, false);
        }
    }

    // Epilogue: bias + ReLU + scatter-add into out[receiver].
    const float bias = b1[col];
    if (e0 + TILE_M <= (long long)nEdges) {
        // Fast path: whole tile valid -> straight-line atomics, no divergence.
        #pragma unroll
        for (int m = 0; m < MSUB; ++m) {
            #pragma unroll
            for (int r = 0; r < 8; ++r) {
                int row = m * 16 + 8 * lhi + r;
                float vmsg = fmaxf(acc[m][r] + bias, 0.f);
                atomicAdd(out + (long long)rIdx[row] * H + col, vmsg);
            }
        }
    } else {
        #pragma unroll
        for (int m = 0; m < MSUB; ++m) {
            #pragma unroll
            for (int r = 0; r < 8; ++r) {
                int row = m * 16 + 8 * lhi + r;
                if (e0 + row < (long long)nEdges) {
                    float vmsg = fmaxf(acc[m][r] + bias, 0.f);
                    atomicAdd(out + (long long)rIdx[row] * H + col, vmsg);
                }
            }
        }
    }
}

// ---------------------------------------------------------------------------
// Kernel 2: in-place LayerNorm over rows of out. One wave32 per row.
// ---------------------------------------------------------------------------
__global__ __launch_bounds__(256)
void ln_kernel(float* __restrict__ out,
               const float* __restrict__ gamma,
               const float* __restrict__ beta,
               int nNodes)
{
    const int wave = threadIdx.x >> 5;
    const int lane = threadIdx.x & 31;
    const long long row = (long long)blockIdx.x * 8 + wave;
    if (row >= nNodes) return;

    float* rowp = out + row * H;
    v4f v = *(const v4f*)(rowp + lane * 4);

    float s  = v.x + v.y + v.z + v.w;
    float sq = v.x * v.x + v.y * v.y + v.z * v.z + v.w * v.w;
    #pragma unroll
    for (int off = 16; off >= 1; off >>= 1) {
        s  += __shfl_xor(s,  off, 32);
        sq += __shfl_xor(sq, off, 32);
    }
    const float mu   = s * (1.0f / H);
    const float var  = sq * (1.0f / H) - mu * mu;
    const float rinv = rsqrtf(var + 1e-5f);

    v4f g  = *(const v4f*)(gamma + lane * 4);
    v4f bb = *(const v4f*)(beta  + lane * 4);
    v4f o;
    o.x = g.x * (v.x - mu) * rinv + bb.x;
    o.y = g.y * (v.y - mu) * rinv + bb.y;
    o.z = g.z * (v.z - mu) * rinv + bb.z;
    o.w = g.w * (v.w - mu) * rinv + bb.w;
    *(v4f*)(rowp + lane * 4) = o;
}

// ---------------------------------------------------------------------------
extern "C" void kernel_launch(void* const* d_in, const int* in_sizes, int n_in,
                              void* d_out, int out_size, void* d_ws, size_t ws_size,
                              hipStream_t stream)
{
    const float* x        = (const float*)d_in[0];
    const int*   senders  = (const int*)  d_in[1];
    const int*   receivers= (const int*)  d_in[2];
    const float* ef       = (const float*)d_in[3];
    const float* W1       = (const float*)d_in[4];
    const float* b1       = (const float*)d_in[5];
    const float* gamma    = (const float*)d_in[6];
    const float* beta     = (const float*)d_in[7];

    const int nNodes = in_sizes[0] / H;
    const int nEdges = in_sizes[1];
    float* out = (float*)d_out;

    // Seed out with x: residual is pre-added, messages accumulate on top.
    hipMemcpyAsync(out, x, (size_t)nNodes * H * sizeof(float),
                   hipMemcpyDeviceToDevice, stream);

    const int gemmBlocks = (nEdges + TILE_M - 1) / TILE_M;
    edge_msg_kernel<<<gemmBlocks, 256, 0, stream>>>(x, senders, receivers, ef,
                                                    W1, b1, out, nEdges);

    const int lnBlocks = (nNodes + 7) / 8;
    ln_kernel<<<lnBlocks, 256, 0, stream>>>(out, gamma, beta, nNodes);
}